// Model_84250078478347
// MI455X (gfx1250) — compile-verified
//
#include <hip/hip_runtime.h>
#include <stdint.h>

#define NB   128        // batches
#define NE   16         // experts / rows summed
#define NS   32768      // sequence length
#define TJ   1024       // output positions per block
#define TPB  256        // threads per block (8 wave32)
#define CH   (TJ / TPB) // 4 positions per thread

#define AS1 __attribute__((address_space(1)))
#define AS3 __attribute__((address_space(3)))

#if defined(__AMDGCN__) && __has_builtin(__builtin_amdgcn_global_load_async_to_lds_b32)
#define USE_ASYNC 1
#else
#define USE_ASYNC 0
#endif

// Monotone float <-> uint key so atomicMax(uint) implements float max (incl. negatives).
__device__ __forceinline__ unsigned fkey(float f) {
    unsigned u = __float_as_uint(f);
    return u ^ ((u & 0x80000000u) ? 0xFFFFFFFFu : 0x80000000u);
}
__device__ __forceinline__ float funkey(unsigned k) {
    unsigned u = k ^ ((k & 0x80000000u) ? 0x80000000u : 0xFFFFFFFFu);
    return __uint_as_float(u);
}

__global__ void init_maxkey(unsigned* __restrict__ mk) {
    if (threadIdx.x < NB) mk[threadIdx.x] = 0u;  // key(-NaN) == ordering minimum
}

__global__ __launch_bounds__(TPB) void scatter_sum_kernel(
    const float* __restrict__ events,   // [NB, NE, NS]
    const int*   __restrict__ indices,  // [NB, NE]
    float*       __restrict__ out,      // [NB, NS] raw sums
    unsigned*    __restrict__ maxkey)   // [NB]
{
#if USE_ASYNC
    __shared__ float smem[NE * TJ];     // 64 KiB staging tile
#endif
    __shared__ float wmax[TPB / 32];

    const int b   = blockIdx.y;
    const int j0  = blockIdx.x * TJ;
    const int tid = threadIdx.x;

    int idx[NE];
#pragma unroll
    for (int e = 0; e < NE; ++e) idx[e] = indices[b * NE + e];  // uniform -> SGPR

    const float* ev_b = events + (size_t)b * NE * NS;

#if USE_ASYNC
    // Stage 16 shifted row segments into LDS via async DMA (ASYNCcnt path).
    // Global address clamped to row start for j < idx_e; masked at consume time.
#pragma unroll
    for (int e = 0; e < NE; ++e) {
        const float* row = ev_b + (size_t)e * NS;
#pragma unroll
        for (int c = 0; c < CH; ++c) {
            const int j = j0 + c * TPB + tid;
            int d = j - idx[e];
            d = d < 0 ? 0 : d;
            __builtin_amdgcn_global_load_async_to_lds_b32(
                (AS1 int*)(row + d),
                (AS3 int*)(&smem[e * TJ + c * TPB + tid]),
                0, 0);
        }
    }
#if __has_builtin(__builtin_amdgcn_s_wait_asynccnt)
    __builtin_amdgcn_s_wait_asynccnt(0);
#else
    asm volatile("s_wait_asynccnt 0" ::: "memory");
#endif
    // Each wave consumes exactly the LDS words it staged: no barrier required.
#endif

    float m = -__builtin_inff();
#pragma unroll
    for (int c = 0; c < CH; ++c) {
        const int j = j0 + c * TPB + tid;
        float a = 0.f;
#pragma unroll
        for (int e = 0; e < NE; ++e) {
            const int d = j - idx[e];
#if USE_ASYNC
            const float v = smem[e * TJ + c * TPB + tid];
#else
            const float v = ev_b[(size_t)e * NS + (d < 0 ? 0 : d)];
#endif
            a += (d >= 0) ? v : 0.f;
        }
        out[(size_t)b * NS + j] = a;
        m = fmaxf(m, a);
    }

    // wave32 max reduction, then one atomic per block
#pragma unroll
    for (int off = 16; off > 0; off >>= 1)
        m = fmaxf(m, __shfl_xor(m, off, 32));
    if ((tid & 31) == 0) wmax[tid >> 5] = m;
    __syncthreads();
    if (tid == 0) {
        float bm = wmax[0];
#pragma unroll
        for (int w = 1; w < TPB / 32; ++w) bm = fmaxf(bm, wmax[w]);
        atomicMax(&maxkey[b], fkey(bm));
    }
}

__global__ __launch_bounds__(TPB) void normalize_kernel(
    float* __restrict__ out, const unsigned* __restrict__ maxkey)
{
    const int b = blockIdx.y;
    const float denom = funkey(maxkey[b]) + 1e-8f;
    const int base = blockIdx.x * (TPB * 4) + threadIdx.x * 4;
    float4* p = (float4*)(out + (size_t)b * NS + base);
    float4 v = *p;
    v.x /= denom; v.y /= denom; v.z /= denom; v.w /= denom;
    *p = v;
}

extern "C" void kernel_launch(void* const* d_in, const int* in_sizes, int n_in,
                              void* d_out, int out_size, void* d_ws, size_t ws_size,
                              hipStream_t stream) {
    const float* events  = (const float*)d_in[0];
    const int*   indices = (const int*)d_in[1];   // harness passes integer inputs as int32
    float*       out     = (float*)d_out;         // [128, 1, 32768] f32
    unsigned*    maxkey  = (unsigned*)d_ws;       // 128 * 4 bytes of scratch

    init_maxkey<<<dim3(1), dim3(128), 0, stream>>>(maxkey);

    dim3 grid1(NS / TJ, NB);                      // (32, 128)
    scatter_sum_kernel<<<grid1, TPB, 0, stream>>>(events, indices, out, maxkey);

    dim3 grid2(NS / (TPB * 4), NB);               // (32, 128)
    normalize_kernel<<<grid2, TPB, 0, stream>>>(out, maxkey);
}